// SymmetricKLDLoss_22076131902015
// MI455X (gfx1250) — compile-verified
//
#include <hip/hip_runtime.h>
#include <hip/hip_bf16.h>

// SymmetricKLDLoss (shape_only): per-element 2x2 closed-form symmetric KL.
// Memory-bound: 288MB read + 16MB write -> ~13us floor @ 23.3 TB/s.
// Data path: gfx1250 async global->LDS DMA (ASYNCcnt), per-wave double buffer,
// NT temporal hints to stream past the 192MB L2. No GEMM structure -> no WMMA.

#define EPSV            1e-7f
#define FLT_MAX_C       3.4028234663852886e+38f
#define WAVES_PER_BLOCK 4
#define BLOCK_THREADS   (WAVES_PER_BLOCK * 32)
#define TILE_ELEMS      128                   // elements per wave per stage
#define TILE_BYTES      (TILE_ELEMS * 36)     // 4608 B = 9 x (32 lanes x 16 B)
#define TILE_FLOATS     (TILE_ELEMS * 9)      // 1152 floats
#define J_LOADS         9                     // b128 async loads per lane per array
#define ASYNC_PER_TILE  (2 * J_LOADS)         // 18 instrs -> ASYNCcnt +18 per stage

#define WAIT_ASYNC(n) asm volatile("s_wait_asynccnt %0" :: "i"(n) : "memory")

// Issue one wave-tile (128 elements of both arrays) into this wave's LDS slice.
// Each lane moves 9 x 16B chunks per array via the async DMA path (GV mode).
__device__ __forceinline__ void issue_tile(const char* a1b, const char* a2b,
                                           const float* l1, const float* l2,
                                           long long tile, int lane) {
    unsigned long long g1 = (unsigned long long)(size_t)(a1b) +
                            (unsigned long long)tile * TILE_BYTES + (unsigned)(lane * 16);
    unsigned long long g2 = (unsigned long long)(size_t)(a2b) +
                            (unsigned long long)tile * TILE_BYTES + (unsigned)(lane * 16);
    unsigned d1 = (unsigned)(size_t)l1 + (unsigned)(lane * 16);   // low 32 bits of generic
    unsigned d2 = (unsigned)(size_t)l2 + (unsigned)(lane * 16);   // LDS addr == ds offset
#pragma unroll
    for (int j = 0; j < J_LOADS; ++j) {
        asm volatile("global_load_async_to_lds_b128 %0, %1, off th:TH_LOAD_NT"
                     :: "v"(d1 + (unsigned)(j * 512)),
                        "v"(g1 + (unsigned long long)(j * 512)) : "memory");
        asm volatile("global_load_async_to_lds_b128 %0, %1, off th:TH_LOAD_NT"
                     :: "v"(d2 + (unsigned)(j * 512)),
                        "v"(g2 + (unsigned long long)(j * 512)) : "memory");
    }
}

// jnp.nan_to_num(x, nan=10.0): NaN->10, +Inf->FLT_MAX, -Inf->-FLT_MAX
__device__ __forceinline__ float fixup(float x) {
    if (__builtin_isnan(x)) return 10.0f;
    return fminf(fmaxf(x, -FLT_MAX_C), FLT_MAX_C);
}

__device__ __forceinline__ float sym_kl(float m00a, float m01a, float m10a, float m11a,
                                        float m00b, float m01b, float m10b, float m11b) {
    float a1 = m00a + EPSV, b1 = m01a, c1 = m10a, d1 = m11a + EPSV;
    float a2 = m00b + EPSV, b2 = m01b, c2 = m10b, d2 = m11b + EPSV;
    float det1 = a1 * d1 - b1 * c1;
    float det2 = a2 * d2 - b2 * c2;
    // tr(inv(cov2)@cov1) and tr(inv(cov1)@cov2) share the same numerator.
    float num  = a1 * d2 + a2 * d1 - b2 * c1 - c2 * b1;
    float tr12 = num / det2;                       // raw det, matching adj/det math
    float tr21 = num / det1;
    float lg   = __logf(fmaxf(det2, EPSV)) - __logf(fmaxf(det1, EPSV));
    float kl12 = 0.5f * (tr12 - 2.0f + lg);
    float kl21 = 0.5f * (tr21 - 2.0f - lg);
    return 0.5f * (fixup(kl12) + fixup(kl21));
}

__global__ void SymmetricKLDLoss_22076131902015_kernel(const float* __restrict__ A1,
                                                       const float* __restrict__ A2,
                                                       float* __restrict__ out,
                                                       long long N) {
    // [double-buffer][wave][array][tile floats] = 73,728 B per workgroup
    __shared__ float lds[2][WAVES_PER_BLOCK][2][TILE_FLOATS];

    const int lane = threadIdx.x & 31;
    const int wave = threadIdx.x >> 5;
    const long long gwave  = (long long)blockIdx.x * WAVES_PER_BLOCK + wave;
    const long long nwaves = (long long)gridDim.x * WAVES_PER_BLOCK;
    const long long tiles  = N / TILE_ELEMS;

    const char* a1b = (const char*)A1;
    const char* a2b = (const char*)A2;

    long long t  = gwave;
    int       buf = 0;
    if (t < tiles)
        issue_tile(a1b, a2b, &lds[0][wave][0][0], &lds[0][wave][1][0], t, lane);

    while (t < tiles) {
        long long tn = t + nwaves;
        if (tn < tiles) {
            issue_tile(a1b, a2b, &lds[buf ^ 1][wave][0][0], &lds[buf ^ 1][wave][1][0], tn, lane);
            WAIT_ASYNC(ASYNC_PER_TILE);   // oldest 18 retired (in-order completion)
        } else {
            WAIT_ASYNC(0);
        }
        const float* l1 = &lds[buf][wave][0][0];
        const float* l2 = &lds[buf][wave][1][0];
#pragma unroll
        for (int e = 0; e < TILE_ELEMS / 32; ++e) {
            int li = e * 32 + lane;               // stride-9 float offsets: bank-conflict-free
            int o  = 9 * li;
            float r = sym_kl(l1[o], l1[o + 1], l1[o + 3], l1[o + 4],
                             l2[o], l2[o + 1], l2[o + 3], l2[o + 4]);
            __builtin_nontemporal_store(r, &out[t * TILE_ELEMS + li]);
        }
        buf ^= 1;
        t = tn;
    }

    // Tail: N % 128 elements via direct guarded loads (N=4M -> empty, kept generic)
    long long base     = tiles * TILE_ELEMS;
    long long gtid     = (long long)blockIdx.x * blockDim.x + threadIdx.x;
    long long nthreads = (long long)gridDim.x * blockDim.x;
    for (long long i = base + gtid; i < N; i += nthreads) {
        const float* p1 = A1 + i * 9;
        const float* p2 = A2 + i * 9;
        out[i] = sym_kl(p1[0], p1[1], p1[3], p1[4],
                        p2[0], p2[1], p2[3], p2[4]);
    }
}

extern "C" void kernel_launch(void* const* d_in, const int* in_sizes, int n_in,
                              void* d_out, int out_size, void* d_ws, size_t ws_size,
                              hipStream_t stream) {
    (void)n_in; (void)d_ws; (void)ws_size; (void)out_size;
    const float* A1 = (const float*)d_in[0];
    const float* A2 = (const float*)d_in[1];
    float* out = (float*)d_out;

    long long N     = (long long)in_sizes[0] / 9;   // (N,3,3) float32
    long long tiles = N / TILE_ELEMS;
    long long want  = (tiles + WAVES_PER_BLOCK - 1) / WAVES_PER_BLOCK;
    int blocks = (int)(want < 1 ? 1 : (want > 2048 ? 2048 : want));

    SymmetricKLDLoss_22076131902015_kernel<<<blocks, BLOCK_THREADS, 0, stream>>>(A1, A2, out, N);
}